// GCN_24481313587812
// MI455X (gfx1250) — compile-verified
//
#include <hip/hip_runtime.h>

typedef float v2f __attribute__((ext_vector_type(2)));
typedef float v8f __attribute__((ext_vector_type(8)));

#define N_NODES 50000
#define N_EDGES 800000
#define D_HIDF  96
#define D_OUTF  40
#define BN_EPS  1e-5f

// ---------------------------------------------------------------- utilities
__global__ void fill_zero_kernel(float* __restrict__ p, int n) {
    int i = blockIdx.x * blockDim.x + threadIdx.x;
    if (i < n) p[i] = 0.0f;
}

__global__ void degree_accum_kernel(const int* __restrict__ dst,
                                    float* __restrict__ deg, int e) {
    int i = blockIdx.x * blockDim.x + threadIdx.x;
    if (i < e) atomicAdd(&deg[dst[i]], 1.0f);
}

__global__ void make_dinv_kernel(float* __restrict__ deg, int n) {
    int i = blockIdx.x * blockDim.x + threadIdx.x;
    if (i < n) deg[i] = rsqrtf(deg[i] + 1.0f);   // deg includes self-loop
}

// ------------------------------------------------------- fp32 WMMA GEMM
// out[nrows x NACT] = A[nrows x 96] * W[96 x NACT]   (row-major)
// One wave32 computes a 16 x NPAD strip (NPAD/16 WMMA tiles), reusing each
// A fragment across all col tiles. W is staged in LDS once per block,
// zero-padded to NPAD cols, packed as (k,k+1) pairs so B fragments are
// single ds_load_b64 with immediate offsets.
// Fragment layouts (ISA 7.12.2), lane m=lane&15, hi=lane>>4:
//   A 16x4 : VGPR v <-> K = k0 + 2*hi + v      (one b64 load)
//   B 4x16 : same K mapping, col n             (one b64 LDS load)
//   C 16x16: VGPR r <-> row r + 8*hi, col n
template <int NACT, int NPAD>
__global__ __launch_bounds__(256)
void gemm_wmma_f32_kernel(const float* __restrict__ A,
                          const float* __restrict__ W,
                          float* __restrict__ out, int nrows) {
    constexpr int CT = NPAD / 16;          // col tiles per strip
    __shared__ float ldsW[96 * NPAD];      // pair-packed: [(k>>1)*2*NPAD + 2*n + (k&1)]

    const int tid = threadIdx.x;
    for (int idx = tid; idx < 96 * NPAD; idx += 256) {
        int k = idx / NPAD;
        int n = idx - k * NPAD;
        float v = (n < NACT) ? W[k * NACT + n] : 0.0f;
        ldsW[(k >> 1) * (2 * NPAD) + 2 * n + (k & 1)] = v;
    }
    __syncthreads();

    const int wave = tid >> 5;
    const int lane = tid & 31;
    const int row_tile = blockIdx.x * 8 + wave;
    if (row_tile * 16 >= nrows) return;    // wave-uniform (after barrier)

    const int m  = lane & 15;
    const int hi = lane >> 4;

    // A: pair pointer into this lane's row; pair index = k0/2 + hi
    const v2f* ap = (const v2f*)(A + (size_t)(row_tile * 16 + m) * 96);
    // B: pair pointer; pair index = (k0/2 + hi)*NPAD + (t*16 + m)
    const v2f* wb = (const v2f*)ldsW + hi * NPAD + m;

    v8f acc[CT];
#pragma unroll
    for (int t = 0; t < CT; ++t) acc[t] = v8f{};

#pragma unroll
    for (int k0 = 0; k0 < 96; k0 += 4) {
        v2f a = ap[(k0 >> 1) + hi];
#pragma unroll
        for (int t = 0; t < CT; ++t) {
            v2f b = wb[(k0 >> 1) * NPAD + t * 16];
            acc[t] = __builtin_amdgcn_wmma_f32_16x16x4_f32(
                false, a, false, b, (short)0, acc[t], false, false);
        }
    }

    const int row0 = row_tile * 16 + 8 * hi;
#pragma unroll
    for (int t = 0; t < CT; ++t) {
        int n = t * 16 + m;
        if (NACT == NPAD || n < NACT) {
            float* o = out + (size_t)row0 * NACT + n;
#pragma unroll
            for (int r = 0; r < 8; ++r)
                o[(size_t)r * NACT] = acc[t][r];
        }
    }
}

// --------------------------------------------- self-loop init (also zeroes agg)
template <int D>
__global__ void selfloop_init_kernel(const float* __restrict__ hw,
                                     const float* __restrict__ dinv,
                                     float* __restrict__ agg, int n) {
    int i = blockIdx.x * blockDim.x + threadIdx.x;
    if (i >= n * D) return;
    float di = dinv[i / D];
    agg[i] = hw[i] * di * di;
}

// ------------------------------------------------------------- edge scatter
template <int D>
__global__ void edge_scatter_kernel(const int* __restrict__ src,
                                    const int* __restrict__ dst,
                                    const float* __restrict__ dinv,
                                    const float* __restrict__ hw,
                                    float* __restrict__ agg, int e) {
    int i = blockIdx.x * blockDim.x + threadIdx.x;
    if (i >= e * D) return;                 // 800k*96 = 76.8M < 2^31
    int ed = i / D;
    int f  = i - ed * D;
    int s  = src[ed];
    int t  = dst[ed];
    float coef = dinv[s] * dinv[t];
    atomicAdd(&agg[t * D + f], hw[s * D + f] * coef);
}

// ---------------------------------------------------- epilogues
template <int D>
__global__ void bias_bn_relu_kernel(const float* __restrict__ agg,
                                    const float* __restrict__ b,
                                    const float* __restrict__ g,
                                    const float* __restrict__ beta,
                                    const float* __restrict__ rm,
                                    const float* __restrict__ rv,
                                    float* __restrict__ out, int n) {
    int i = blockIdx.x * blockDim.x + threadIdx.x;
    if (i >= n * D) return;
    int f = i % D;
    float v = agg[i] + b[f];
    v = (v - rm[f]) * rsqrtf(rv[f] + BN_EPS) * g[f] + beta[f];
    out[i] = v > 0.0f ? v : 0.0f;
}

template <int D>
__global__ void bias_only_kernel(const float* __restrict__ agg,
                                 const float* __restrict__ b,
                                 float* __restrict__ out, int n) {
    int i = blockIdx.x * blockDim.x + threadIdx.x;
    if (i >= n * D) return;
    out[i] = agg[i] + b[i % D];
}

// ---------------------------------------------------------------- launcher
static inline int blocks_for(long long work, int t) {
    return (int)((work + t - 1) / t);
}

template <int DOUT, int DPAD>
static void run_gcn_layer(const float* hin, const float* W, const float* b,
                          const float* g, const float* beta,
                          const float* rm, const float* rv,
                          const int* src, const int* dst, const float* dinv,
                          float* hw, float* agg, float* hout, hipStream_t stream) {
    const int T = 256;
    int row_tiles = N_NODES / 16;                       // 3125 (exact)
    gemm_wmma_f32_kernel<DOUT, DPAD>
        <<<blocks_for(row_tiles, 8), T, 0, stream>>>(hin, W, hw, N_NODES);
    selfloop_init_kernel<DOUT>
        <<<blocks_for((long long)N_NODES * DOUT, T), T, 0, stream>>>(hw, dinv, agg, N_NODES);
    edge_scatter_kernel<DOUT>
        <<<blocks_for((long long)N_EDGES * DOUT, T), T, 0, stream>>>(src, dst, dinv, hw, agg, N_EDGES);
    if (g) {
        bias_bn_relu_kernel<DOUT>
            <<<blocks_for((long long)N_NODES * DOUT, T), T, 0, stream>>>(
                agg, b, g, beta, rm, rv, hout, N_NODES);
    } else {
        bias_only_kernel<DOUT>
            <<<blocks_for((long long)N_NODES * DOUT, T), T, 0, stream>>>(agg, b, hout, N_NODES);
    }
}

extern "C" void kernel_launch(void* const* d_in, const int* in_sizes, int n_in,
                              void* d_out, int out_size, void* d_ws, size_t ws_size,
                              hipStream_t stream) {
    const float* x     = (const float*)d_in[0];
    const int*   ei    = (const int*)d_in[1];
    const float* W1    = (const float*)d_in[2];
    const float* b1    = (const float*)d_in[3];
    const float* g1    = (const float*)d_in[4];
    const float* beta1 = (const float*)d_in[5];
    const float* rm1   = (const float*)d_in[6];
    const float* rv1   = (const float*)d_in[7];
    const float* W2    = (const float*)d_in[8];
    const float* b2    = (const float*)d_in[9];
    const float* g2    = (const float*)d_in[10];
    const float* beta2 = (const float*)d_in[11];
    const float* rm2   = (const float*)d_in[12];
    const float* rv2   = (const float*)d_in[13];
    const float* W3    = (const float*)d_in[14];
    const float* b3    = (const float*)d_in[15];

    const int* src = ei;            // edge_index[0]
    const int* dst = ei + N_EDGES;  // edge_index[1]

    float* ws   = (float*)d_ws;
    float* dinv = ws;                                    // N
    float* hw   = dinv + N_NODES;                        // N * 96
    float* agg  = hw   + (size_t)N_NODES * D_HIDF;       // N * 96
    float* h    = agg  + (size_t)N_NODES * D_HIDF;       // N * 96
    float* out  = (float*)d_out;

    const int T = 256;

    // ---- symmetric-norm coefficients: dinv = (deg+1)^-1/2
    fill_zero_kernel<<<blocks_for(N_NODES, T), T, 0, stream>>>(dinv, N_NODES);
    degree_accum_kernel<<<blocks_for(N_EDGES, T), T, 0, stream>>>(dst, dinv, N_EDGES);
    make_dinv_kernel<<<blocks_for(N_NODES, T), T, 0, stream>>>(dinv, N_NODES);

    // ---- three GCN layers
    run_gcn_layer<96, 96>(x, W1, b1, g1, beta1, rm1, rv1,
                          src, dst, dinv, hw, agg, h, stream);
    run_gcn_layer<96, 96>(h, W2, b2, g2, beta2, rm2, rv2,
                          src, dst, dinv, hw, agg, h, stream);
    run_gcn_layer<40, 48>(h, W3, b3, nullptr, nullptr, nullptr, nullptr,
                          src, dst, dinv, hw, agg, out, stream);
}